// ConvInsert_22127671509397
// MI455X (gfx1250) — compile-verified
//
#include <hip/hip_runtime.h>

// ConvInsert on MI455X (gfx1250, wave32).
// Memory-bound: ~302 MB total traffic -> ~13us floor at 23.3 TB/s.
// Dot products via V_WMMA_F32_16X16X4_F32 (fp32 WMMA, K=16 accumulated in 4 steps).
// B fragments precomputed branchlessly once per wave; result tail is branchless
// (clamped bias index + address-selected store into a d_ws sink for the single
// overflow slot) so no per-element exec-mask blocks remain.

typedef float v2f __attribute__((ext_vector_type(2)));
typedef float v8f __attribute__((ext_vector_type(8)));

#define IN_N    4096   // floats per row
#define WINDOWS 511    // overlapping 16-wide windows per row
#define OUT_N   5118   // 511*10 + 8 floats per output row

__global__ __launch_bounds__(256)
void convinsert_wmma_kernel(const float* __restrict__ in,
                            const float* __restrict__ w1,
                            const float* __restrict__ w2,
                            const float* __restrict__ bias,  // 1022 floats, b[2w+t]
                            float* __restrict__ out,
                            float* __restrict__ ws_sink)     // discard target for w==511
{
    const int row = blockIdx.x;
    const float* in_row  = in  + (size_t)row * IN_N;
    float*       out_row = out + (size_t)row * OUT_N;
    const int tid = threadIdx.x;

    // ---------------- Phase A: stream-copy slices into strided output slots.
    // out[slice*10 + j] = in[slice*8 + j], j in 0..7 (holds for the final slice:
    // 511*10 = 5110). 1024 float4 loads (16B aligned); stores expressed as float2
    // (8B aligned) — backend merges them into b128 stores.
#pragma unroll
    for (int it = 0; it < 4; ++it) {
        const int g = tid + it * 256;              // 0..1023: g-th float4 of the row
        const float4 v = *(const float4*)(in_row + g * 4);
        const int slice = g >> 1;
        const int j     = (g & 1) * 4;
        float* dst = out_row + slice * 10 + j;
        ((float2*)dst)[0] = make_float2(v.x, v.y);
        ((float2*)dst)[1] = make_float2(v.z, v.w);
    }

    // ---------------- Phase B: outs[w][t] = dot(window_w, W[:,t]) + b[2w+t]
    // via V_WMMA_F32_16X16X4_F32. Per wave: 4 tiles of 16 windows, each tile = 4
    // accumulating WMMAs over K=16. 8 waves * 4 tiles * 16 = 512 >= 511 windows.
    const int lane   = tid & 31;
    const int wave   = tid >> 5;
    const int m      = lane & 15;          // A row / B column held by this lane
    const bool lo    = lane < 16;
    const int ksub   = lo ? 0 : 2;         // K sub-index per the f32 A/B VGPR layout

    // Precompute B fragments (lane-invariant across tiles). Column n=m of
    // W = [w1 | w2 | 0 ...]: select branchlessly with {0,1} multiplier masks so
    // the w1/w2 loads stay unconditional (no exec-masked load blocks).
    const float fsel0 = (m == 0) ? 1.0f : 0.0f;
    const float fsel1 = (m == 1) ? 1.0f : 0.0f;
    v2f bfrag[4];
#pragma unroll
    for (int s = 0; s < 4; ++s) {
        const int kb = 4 * s + ksub;                   // even -> 8B aligned
        const float2 p = *(const float2*)(w1 + kb);
        const float2 q = *(const float2*)(w2 + kb);
        bfrag[s].x = p.x * fsel0 + q.x * fsel1;
        bfrag[s].y = p.y * fsel0 + q.y * fsel1;
    }

#pragma unroll
    for (int t2 = 0; t2 < 4; ++t2) {
        const int wbase = wave * 64 + t2 * 16;              // first window of tile
        const int wc    = min(wbase + m, WINDOWS - 1);      // clamp A addr (sink-guarded)
        const float* aptr = in_row + wc * 8 + ksub;         // step loads use imm offsets

        v8f acc = {0.f, 0.f, 0.f, 0.f, 0.f, 0.f, 0.f, 0.f};

#pragma unroll
        for (int s = 0; s < 4; ++s) {
            // A fragment: window row m, K = 4s+ksub, 4s+ksub+1 (8B-aligned float2).
            const float2 a2 = *(const float2*)(aptr + 4 * s);
            v2f av; av.x = a2.x; av.y = a2.y;
            // 8 args: (neg_a, A, neg_b, B, c_mod, C, reuse_a, reuse_b)
            acc = __builtin_amdgcn_wmma_f32_16x16x4_f32(
                false, av, false, bfrag[s], (short)0, acc, false, false);
        }

        // D layout: VGPR r holds M=r (lanes 0-15) / M=8+r (lanes 16-31), N=lane&15.
        // Columns N<2 carry the two dot products for windows wbase+M.
        // Branchless tail: only w==511 is invalid -> clamp the bias index and
        // redirect that single store to the scratch sink (address select, no branch).
        if (m < 2) {
            float* sink = ws_sink + tid;
#pragma unroll
            for (int r = 0; r < 8; ++r) {
                const int mm  = lo ? r : (8 + r);
                const int w   = wbase + mm;
                const int wcl = min(w, WINDOWS - 1);
                const float v = acc[r] + bias[2 * wcl + m];
                float* p = (w < WINDOWS) ? (out_row + w * 10 + 8 + m) : sink;
                *p = v;
            }
        }
    }
}

extern "C" void kernel_launch(void* const* d_in, const int* in_sizes, int n_in,
                              void* d_out, int out_size, void* d_ws, size_t ws_size,
                              hipStream_t stream) {
    (void)n_in; (void)out_size; (void)ws_size;
    const float* in  = (const float*)d_in[0];
    const float* w1  = (const float*)d_in[1];
    const float* w2  = (const float*)d_in[2];
    const float* b   = (const float*)d_in[3];
    float* out = (float*)d_out;
    float* ws  = (float*)d_ws;

    const int rows = in_sizes[0] / IN_N;   // 32*256 = 8192
    convinsert_wmma_kernel<<<rows, 256, 0, stream>>>(in, w1, w2, b, out, ws);
}